// VectorQuantizer_OCB_bk_35639638622555
// MI455X (gfx1250) — compile-verified
//
#include <hip/hip_runtime.h>
#include <math.h>

// ---- problem constants -----------------------------------------------------
#define NBAT 8
#define NCH  4
#define HH   96
#define WW   96
#define NSITE (NBAT*HH*WW)     // 73728 quantization sites
#define EDIM  32               // embedding dim (= NCH * 8)
#define NE    4096             // codebook size
#define NELEM (NSITE*EDIM)     // 2359296 elements in zz / z_q

#define STAGE_CODES 64                 // codes staged in LDS per barrier
#define NSTAGE (NE / STAGE_CODES)      // 64 stages

typedef __attribute__((ext_vector_type(2))) float v2f;
typedef __attribute__((ext_vector_type(4))) float v4f;
typedef __attribute__((ext_vector_type(8))) float v8f;
typedef __attribute__((ext_vector_type(4))) int   v4i;

// ---- CDNA5 async global->LDS staging (guarded: falls back to sync copy) ----
#if defined(__has_builtin)
#if __has_builtin(__builtin_amdgcn_global_load_async_to_lds_b128) && \
    __has_builtin(__builtin_amdgcn_s_wait_asynccnt)
#define VQ_ASYNC 1
#endif
#endif
#ifndef VQ_ASYNC
#define VQ_ASYNC 0
#endif

#if VQ_ASYNC
__device__ __forceinline__ void vq_async_b128(const float* gsrc, float* ldst) {
  // Builtin signature (from hipcc diagnostic): param0 is a global (AS1)
  // pointer to a 16-byte int vector; param1 the LDS (AS3) counterpart.
  // generic->AS1 is identity; a generic LDS pointer's low 32 bits are the
  // LDS offset, so integer truncation to a 32-bit AS3 pointer is exact.
  __builtin_amdgcn_global_load_async_to_lds_b128(
      (__attribute__((address_space(1))) v4i*)(unsigned long long)gsrc,
      (__attribute__((address_space(3))) v4i*)(unsigned int)(unsigned long long)ldst,
      0, 0);
}
#endif

// ---------------------------------------------------------------------------
// Kernel 1: build zf[N][32] = the pixelshuffled trilinear-upsampled z.
// Depth duplicates (half-pixel resize 1->2 clamps both taps to plane 0), so
// channels c*8+{0..3} == c*8+{4..7}. Each thread handles one (site, c).
// ---------------------------------------------------------------------------
__global__ __launch_bounds__(256) void vq_prep_zf(const float* __restrict__ z,
                                                  float* __restrict__ zf) {
  int t = blockIdx.x * 256 + threadIdx.x;           // 0 .. NSITE*NCH-1
  if (t >= NSITE * NCH) return;
  int n = t >> 2;
  int c = t & 3;
  int b  = n / (HH * WW);
  int r  = n - b * HH * WW;
  int hb = r / WW;
  int wb = r - hb * WW;
  const float* zp = z + (size_t)(b * NCH + c) * HH * WW;
  int hm = hb > 0 ? hb - 1 : 0, hp = hb < HH - 1 ? hb + 1 : HH - 1;
  int wm = wb > 0 ? wb - 1 : 0, wp = wb < WW - 1 ? wb + 1 : WW - 1;
  float m_mm = zp[hm*WW+wm], m_m0 = zp[hm*WW+wb], m_mp = zp[hm*WW+wp];
  float m_0m = zp[hb*WW+wm], m_00 = zp[hb*WW+wb], m_0p = zp[hb*WW+wp];
  float m_pm = zp[hp*WW+wm], m_p0 = zp[hp*WW+wb], m_pp = zp[hp*WW+wp];
  // horizontal half-pixel interp (weights 0.25/0.75; clamp == renormalize)
  float hA0 = 0.25f*m_mm + 0.75f*m_m0;
  float hA1 = 0.75f*m_m0 + 0.25f*m_mp;
  float hB0 = 0.25f*m_0m + 0.75f*m_00;
  float hB1 = 0.75f*m_00 + 0.25f*m_0p;
  float hC0 = 0.25f*m_pm + 0.75f*m_p0;
  float hC1 = 0.75f*m_p0 + 0.25f*m_pp;
  // vertical
  float v00 = 0.25f*hA0 + 0.75f*hB0;   // (s_h=0, s_w=0)
  float v01 = 0.25f*hA1 + 0.75f*hB1;   // (0,1)
  float v10 = 0.75f*hB0 + 0.25f*hC0;   // (1,0)
  float v11 = 0.75f*hB1 + 0.25f*hC1;   // (1,1)
  float* dst = zf + (size_t)n * EDIM + c * 8;
  v4f q = {v00, v01, v10, v11};
  *(v4f*)dst       = q;   // s_d = 0
  *(v4f*)(dst + 4) = q;   // s_d = 1 (duplicate)
}

// ---------------------------------------------------------------------------
// Kernel 2: per-code squared norms |e_j|^2.
// ---------------------------------------------------------------------------
__global__ __launch_bounds__(256) void vq_esq(const float* __restrict__ ew,
                                              float* __restrict__ esq) {
  int j = blockIdx.x * 256 + threadIdx.x;
  if (j >= NE) return;
  const float* p = ew + (size_t)j * EDIM;
  float s = 0.f;
  #pragma unroll
  for (int k = 0; k < EDIM; k += 4) {
    v4f q = *(const v4f*)(p + k);
    s += q.x*q.x + q.y*q.y + q.z*q.z + q.w*q.w;
  }
  esq[j] = s;
}

// ---------------------------------------------------------------------------
// Kernel 3: WMMA distance GEMM + fused row argmin.
//   G = zf @ W^T via V_WMMA_F32_16X16X4_F32; d_proxy = |e_j|^2 - 2*G[n,j].
// 8 waves/block, one 16-row tile per wave.  64 codes (8 KB) are staged in LDS
// per barrier and consumed as FOUR independent accumulator chains so
// consecutive WMMAs never RAW-hazard on the same D registers.  With the async
// path, stage s+1's GLOBAL_LOAD_ASYNC_TO_LDS_B128 DMA (ASYNCcnt) overlaps
// stage s's WMMA work, double-buffered, one barrier per stage.
// ---------------------------------------------------------------------------
__global__ __launch_bounds__(256) void vq_wmma_argmin(
    const float* __restrict__ zf, const float* __restrict__ ew,
    const float* __restrict__ esq, int* __restrict__ idx_i,
    float* __restrict__ idx_f) {
#if VQ_ASYNC
  __shared__ float Bs[2][STAGE_CODES * EDIM];   // 2 x 8 KB double buffer
#else
  __shared__ float Bs[1][STAGE_CODES * EDIM];   // 8 KB
#endif
  const int tid  = threadIdx.x;
  const int wave = tid >> 5;
  const int lane = tid & 31;
  const int half = lane >> 4;                   // 0 or 1
  const int l16  = lane & 15;
  const int rowbase = blockIdx.x * 128 + wave * 16;

  // A fragments: 8 K-slabs of 4; lanes 0-15 hold K{0,1}, lanes 16-31 K{2,3}.
  v2f a[8];
  {
    const float* ap = zf + (size_t)(rowbase + l16) * EDIM + half * 2;
    #pragma unroll
    for (int kk = 0; kk < 8; ++kk) a[kk] = *(const v2f*)(ap + kk * 4);
  }

  float bestv[8];
  int   besti[8];
  #pragma unroll
  for (int r = 0; r < 8; ++r) { bestv[r] = 3.4e38f; besti[r] = 0; }

#if VQ_ASYNC
  {  // prologue: stage 0 into buffer 0 (each thread DMAs 32 contiguous bytes)
    const float* src0 = ew + (size_t)tid * 8;
    vq_async_b128(src0,     &Bs[0][tid * 8]);
    vq_async_b128(src0 + 4, &Bs[0][tid * 8 + 4]);
  }
#endif

  for (int s = 0; s < NSTAGE; ++s) {
#if VQ_ASYNC
    const int cur = s & 1;
    __builtin_amdgcn_s_wait_asynccnt(0);   // my DMA into Bs[cur] landed
    __syncthreads();                       // everyone's landed; prev buf free
    if (s + 1 < NSTAGE) {
      const float* nsrc = ew + (size_t)(s + 1) * STAGE_CODES * EDIM + tid * 8;
      vq_async_b128(nsrc,     &Bs[cur ^ 1][tid * 8]);
      vq_async_b128(nsrc + 4, &Bs[cur ^ 1][tid * 8 + 4]);
    }
    const float* bs = Bs[cur];
#else
    __syncthreads();                       // previous stage consumed
    {
      const float* srcp = ew + (size_t)s * STAGE_CODES * EDIM + tid * 8;
      *(v4f*)&Bs[0][tid * 8]     = *(const v4f*)(srcp);
      *(v4f*)&Bs[0][tid * 8 + 4] = *(const v4f*)(srcp + 4);
    }
    __syncthreads();                       // stage ready
    const float* bs = Bs[0];
#endif

    // four independent 16-col sub-tiles -> four independent WMMA chains
    v8f acc0 = {0.f,0.f,0.f,0.f,0.f,0.f,0.f,0.f};
    v8f acc1 = acc0, acc2 = acc0, acc3 = acc0;
    const float* bp = bs + l16 * EDIM + half * 2;
    #pragma unroll
    for (int kk = 0; kk < 8; ++kk) {
      v2f b0 = *(const v2f*)(bp + kk * 4);
      v2f b1 = *(const v2f*)(bp + 16 * EDIM + kk * 4);
      v2f b2 = *(const v2f*)(bp + 32 * EDIM + kk * 4);
      v2f b3 = *(const v2f*)(bp + 48 * EDIM + kk * 4);
      acc0 = __builtin_amdgcn_wmma_f32_16x16x4_f32(false, a[kk], false, b0,
                                                   (short)0, acc0, false, false);
      acc1 = __builtin_amdgcn_wmma_f32_16x16x4_f32(false, a[kk], false, b1,
                                                   (short)0, acc1, false, false);
      acc2 = __builtin_amdgcn_wmma_f32_16x16x4_f32(false, a[kk], false, b2,
                                                   (short)0, acc2, false, false);
      acc3 = __builtin_amdgcn_wmma_f32_16x16x4_f32(false, a[kk], false, b3,
                                                   (short)0, acc3, false, false);
    }

    const int jb = s * STAGE_CODES + l16;
    float e0 = esq[jb];
    float e1 = esq[jb + 16];
    float e2 = esq[jb + 32];
    float e3 = esq[jb + 48];
    #pragma unroll
    for (int r = 0; r < 8; ++r) {
      float d0 = fmaf(-2.f, acc0[r], e0);
      float d1 = fmaf(-2.f, acc1[r], e1);
      float d2 = fmaf(-2.f, acc2[r], e2);
      float d3 = fmaf(-2.f, acc3[r], e3);
      if (d0 < bestv[r]) { bestv[r] = d0; besti[r] = jb; }
      if (d1 < bestv[r]) { bestv[r] = d1; besti[r] = jb + 16; }
      if (d2 < bestv[r]) { bestv[r] = d2; besti[r] = jb + 32; }
      if (d3 < bestv[r]) { bestv[r] = d3; besti[r] = jb + 48; }
    }
  }

  // reduce (min, argmin) across the 16 lanes of each half (wave32-safe:
  // xor masks 1..8 never cross the half boundary).  Ties -> lower index.
  #pragma unroll
  for (int r = 0; r < 8; ++r) {
    float v = bestv[r];
    int   i = besti[r];
    #pragma unroll
    for (int m = 8; m >= 1; m >>= 1) {
      float ov = __shfl_xor(v, m, 32);
      int   oi = __shfl_xor(i, m, 32);
      if (ov < v || (ov == v && oi < i)) { v = ov; i = oi; }
    }
    if (l16 == 0) {
      int g = rowbase + r + half * 8;
      idx_i[g] = i;
      idx_f[g] = (float)i;      // idx output region of d_out (as float)
    }
  }
}

// ---------------------------------------------------------------------------
// Kernel 4: gather z_q = W[idx], emit y = mean over the 8 sub-channels, and
// accumulate deterministic per-block partial sums for the loss:
//   [0]=sum (zq-zz)^2  [1]=sum zq  [2]=sum zz  [3]=sum zq^2  [4]=sum zz^2
//   [5]=sum zq*zz
// ---------------------------------------------------------------------------
__global__ __launch_bounds__(256) void vq_finalize(
    const float* __restrict__ zf, const float* __restrict__ ew,
    const int* __restrict__ idx_i, float* __restrict__ y,
    float* __restrict__ partials) {
  __shared__ float red[256];
  int t = blockIdx.x * 256 + threadIdx.x;     // (site, c), t < NSITE*NCH
  int n = t >> 2;
  int c = t & 3;
  int id = idx_i[n];
  const float* ep = ew + (size_t)id * EDIM + c * 8;
  const float* zp = zf + (size_t)n  * EDIM + c * 8;
  v4f e0 = *(const v4f*)ep, e1 = *(const v4f*)(ep + 4);
  v4f z0 = *(const v4f*)zp, z1 = *(const v4f*)(zp + 4);
  float ev[8] = {e0.x, e0.y, e0.z, e0.w, e1.x, e1.y, e1.z, e1.w};
  float zv[8] = {z0.x, z0.y, z0.z, z0.w, z1.x, z1.y, z1.z, z1.w};

  float se = 0.f;
  #pragma unroll
  for (int k = 0; k < 8; ++k) se += ev[k];
  int b = n / (HH * WW);
  int r = n - b * HH * WW;
  y[(size_t)(b * NCH + c) * HH * WW + r] = se * 0.125f;

  float sdiff = 0.f, sy = 0.f, sxx = 0.f, syy = 0.f, sxy = 0.f;
  #pragma unroll
  for (int k = 0; k < 8; ++k) {
    float dx = ev[k] - zv[k];
    sdiff += dx * dx;
    sy  += zv[k];
    sxx += ev[k] * ev[k];
    syy += zv[k] * zv[k];
    sxy += ev[k] * zv[k];
  }
  float vals[6] = {sdiff, se, sy, sxx, syy, sxy};
  #pragma unroll
  for (int s = 0; s < 6; ++s) {
    red[threadIdx.x] = vals[s];
    __syncthreads();
    for (int off = 128; off > 0; off >>= 1) {
      if (threadIdx.x < off) red[threadIdx.x] += red[threadIdx.x + off];
      __syncthreads();
    }
    if (threadIdx.x == 0) partials[blockIdx.x * 6 + s] = red[0];
    __syncthreads();
  }
}

// ---------------------------------------------------------------------------
// Kernel 5 (single block): final reduction -> scalar loss.
// loss = 1.25*MSE + pearson(z_q, zz) + 0.01 * max column |.|-sum of W.
// ---------------------------------------------------------------------------
__global__ __launch_bounds__(256) void vq_loss(
    const float* __restrict__ partials, int nblk,
    const float* __restrict__ ew, float* __restrict__ loss_out) {
  __shared__ double sred[256];
  __shared__ float  colred[256];
  __shared__ float  colsum[32];
  int tid = threadIdx.x;

  double sums[6] = {0, 0, 0, 0, 0, 0};
  for (int bIdx = tid; bIdx < nblk; bIdx += 256)
    #pragma unroll
    for (int s = 0; s < 6; ++s) sums[s] += (double)partials[bIdx * 6 + s];
  double tot[6];
  #pragma unroll
  for (int s = 0; s < 6; ++s) {
    sred[tid] = sums[s];
    __syncthreads();
    for (int off = 128; off > 0; off >>= 1) {
      if (tid < off) sred[tid] += sred[tid + off];
      __syncthreads();
    }
    tot[s] = sred[0];
    __syncthreads();
  }

  // regularizer: max over 32 columns of sum_rows |W|
  int j = tid & 31, chunk = tid >> 5;             // 8 chunks x 512 rows
  float cs = 0.f;
  for (int rr = chunk * 512; rr < (chunk + 1) * 512; ++rr)
    cs += fabsf(ew[(size_t)rr * EDIM + j]);
  colred[tid] = cs;
  __syncthreads();
  if (tid < 32) {
    float csum = 0.f;
    for (int k = 0; k < 8; ++k) csum += colred[j + 32 * k];
    colsum[tid] = csum;
  }
  __syncthreads();

  if (tid == 0) {
    double ne = (double)NELEM;
    double sdiff = tot[0], sx = tot[1], syv = tot[2];
    double sxx = tot[3], syy = tot[4], sxy = tot[5];
    double cov = sxy - sx * syv / ne;
    double vx  = sxx - sx * sx / ne;
    double vy  = syy - syv * syv / ne;
    double pear = 0.5 + 0.5 * (cov / (sqrt(vx) * sqrt(vy)));
    float mx = 0.f;
    for (int k = 0; k < 32; ++k) mx = fmaxf(mx, colsum[k]);
    double loss = 1.25 * (sdiff / ne) + pear + 0.01 * (double)mx;
    loss_out[0] = (float)loss;
  }
}

// ---------------------------------------------------------------------------
extern "C" void kernel_launch(void* const* d_in, const int* in_sizes, int n_in,
                              void* d_out, int out_size, void* d_ws, size_t ws_size,
                              hipStream_t stream) {
  (void)in_sizes; (void)n_in; (void)out_size; (void)ws_size;
  const float* z  = (const float*)d_in[0];   // [8,4,96,96]
  const float* ew = (const float*)d_in[1];   // [4096,32]

  float* out  = (float*)d_out;
  float* y    = out;                          // 294912 floats
  float* loss = out + (size_t)NSITE * NCH;    // 1 float
  float* idxf = loss + 1;                     // 73728 "ints" stored as float

  // workspace layout (floats): zf | esq | idx_i(int) | partials
  float* ws       = (float*)d_ws;
  float* zf       = ws;                                   // NELEM
  float* esq      = zf + (size_t)NELEM;                   // NE
  int*   idx_i    = (int*)(esq + NE);                     // NSITE
  float* partials = (float*)(idx_i + NSITE);              // 1152*6

  vq_prep_zf    <<<(NSITE * NCH) / 256, 256, 0, stream>>>(z, zf);
  vq_esq        <<<NE / 256,            256, 0, stream>>>(ew, esq);
  vq_wmma_argmin<<<NSITE / 128,         256, 0, stream>>>(zf, ew, esq, idx_i, idxf);
  vq_finalize   <<<(NSITE * NCH) / 256, 256, 0, stream>>>(zf, ew, idx_i, y, partials);
  vq_loss       <<<1,                   256, 0, stream>>>(partials, (NSITE * NCH) / 256, ew, loss);
}